// Node_42064909697303
// MI455X (gfx1250) — compile-verified
//
#include <hip/hip_runtime.h>

#ifndef __has_builtin
#define __has_builtin(x) 0
#endif

#define TREE_DEPTH     10
#define N_INTERNAL     1023        // 2^10 - 1
#define NFEAT          64
#define ROWS_PER_BLOCK 128
#define NTHREADS       128
#define ROW_STRIDE     68          // floats; 272B pitch: 16B aligned, staggers LDS banks

// CDNA5 async global->LDS path (ASYNCcnt-tracked DMA-style copy, no VGPR round trip)
#if defined(__AMDGCN__) && __has_builtin(__builtin_amdgcn_global_load_async_to_lds_b128)
#define USE_ASYNC_LDS 1
#else
#define USE_ASYNC_LDS 0
#endif

typedef int v4i __attribute__((vector_size(16)));
typedef __attribute__((address_space(1))) v4i* as1_v4ip;   // global
typedef __attribute__((address_space(3))) v4i* as3_v4ip;   // LDS

__device__ __forceinline__ void wait_async_zero() {
#if __has_builtin(__builtin_amdgcn_s_wait_asynccnt)
  __builtin_amdgcn_s_wait_asynccnt(0);
#else
  asm volatile("s_wait_asynccnt 0" ::: "memory");
#endif
}

__global__ __launch_bounds__(NTHREADS) void tree_leaf_kernel(
    const float* __restrict__ x,
    const float* __restrict__ thresholds,
    const int*   __restrict__ var_idx,
    int*         __restrict__ out,
    int nrows)
{
  __shared__ float s_rows[ROWS_PER_BLOCK * ROW_STRIDE];  // 34,816 B
  __shared__ float s_thr[N_INTERNAL + 1];                //  4,096 B
  __shared__ int   s_var[N_INTERNAL + 1];                //  4,096 B

  const int t        = threadIdx.x;
  const int rowBase  = blockIdx.x * ROWS_PER_BLOCK;
  const int rowsHere = min(ROWS_PER_BLOCK, nrows - rowBase);

  // ---- Stage node tables into LDS (8 entries per thread) ----
  for (int i = t; i < N_INTERNAL; i += NTHREADS) {
    s_thr[i] = thresholds[i];
    s_var[i] = var_idx[i];
  }

  // ---- Stage this block's 128 rows (32 KB contiguous) into LDS ----
  const float* gsrc = x + (size_t)rowBase * NFEAT;
#pragma unroll
  for (int k = 0; k < (ROWS_PER_BLOCK * NFEAT / 4) / NTHREADS; ++k) {
    const int q   = t + k * NTHREADS;  // float4 index within the tile
    const int row = q >> 4;            // 16 float4 per row
    const int c4  = q & 15;
    if (row < rowsHere) {
      const float* g = gsrc + q * 4;
      float*       l = &s_rows[row * ROW_STRIDE + c4 * 4];
#if USE_ASYNC_LDS
      __builtin_amdgcn_global_load_async_to_lds_b128((as1_v4ip)g, (as3_v4ip)l,
                                                     /*offset=*/0, /*cpol=*/0);
#else
      *(float4*)l = *(const float4*)g;
#endif
    }
  }
#if USE_ASYNC_LDS
  wait_async_zero();  // barrier does NOT order ASYNCcnt ops (ISA ch.5 / ch.15.5)
#endif
  __syncthreads();

  // ---- Depth-10 traversal entirely out of LDS ----
  if (t < rowsHere) {
    const float* myrow = &s_rows[t * ROW_STRIDE];
    int idx = 0;
#pragma unroll
    for (int d = 0; d < TREE_DEPTH; ++d) {
      const float thr  = s_thr[idx];
      const int   col  = s_var[idx];
      const float feat = myrow[col];
      idx = 2 * idx + 1 + (feat >= thr ? 1 : 0);
    }
    out[rowBase + t] = idx - N_INTERNAL;  // DFS leaf id
  }
}

extern "C" void kernel_launch(void* const* d_in, const int* in_sizes, int n_in,
                              void* d_out, int out_size, void* d_ws, size_t ws_size,
                              hipStream_t stream) {
  const float* x   = (const float*)d_in[0];
  const float* thr = (const float*)d_in[1];
  const int*   var = (const int*)d_in[2];
  // d_in[3] is the python scalar `depth` (== 10, baked in at compile time)
  int* out = (int*)d_out;

  const int B    = in_sizes[0] / NFEAT;
  const int grid = (B + ROWS_PER_BLOCK - 1) / ROWS_PER_BLOCK;
  tree_leaf_kernel<<<grid, NTHREADS, 0, stream>>>(x, thr, var, out, B);
}